// GraphModel_31628139168013
// MI455X (gfx1250) — compile-verified
//
#include <hip/hip_runtime.h>
#include <hip/hip_bf16.h>

// Problem constants (from reference)
#define NN 100000   // nodes
#define FF 128      // feature dim
#define BB 1024     // batch
#define SS 32       // samples per hop
#define DD 64       // hidden dim

typedef float v2f_t __attribute__((ext_vector_type(2)));
typedef float v8f_t __attribute__((ext_vector_type(8)));

// V_WMMA_F32_16X16X4_F32: D(16x16 f32) = A(16x4 f32) * B(4x16 f32) + C
// 8-arg pattern: (neg_a, A, neg_b, B, c_mod, C, reuse_a, reuse_b)
#if defined(__AMDGCN__) && __has_builtin(__builtin_amdgcn_wmma_f32_16x16x4_f32)
#define HAVE_WMMA_F32 1
#endif

__device__ __forceinline__ v8f_t wmma_f32_4(v2f_t a, v2f_t b, v8f_t c) {
#ifdef HAVE_WMMA_F32
    return __builtin_amdgcn_wmma_f32_16x16x4_f32(false, a, false, b,
                                                 (short)0, c, false, false);
#else
    // Host-pass / missing-builtin placeholder (never executed on gfx1250).
    (void)a; (void)b;
    return c;
#endif
}

// ---------------------------------------------------------------------------
// Kernel 1: H[N, 64] = relu(features[N,128] @ W1[128,64] + b1)
// Block = 128 threads (4 waves). Each block -> 16 rows of H; each wave -> one
// 16x16 output tile (wave w covers columns w*16..w*16+15). K stepped by 4.
// ---------------------------------------------------------------------------
__global__ void k1_compute_h(const float* __restrict__ feat,
                             const float* __restrict__ W1,
                             const float* __restrict__ b1,
                             float* __restrict__ H) {
    __shared__ float sA[16 * FF];   // 16 feature rows (8 KB)
    __shared__ float sW[FF * DD];   // W1 (32 KB)
    const int t    = threadIdx.x;
    const int row0 = blockIdx.x * 16;

    // features rows row0..row0+15 are contiguous in memory -> coalesced copy
    for (int idx = t; idx < 16 * FF; idx += 128)
        sA[idx] = feat[(size_t)row0 * FF + idx];
    for (int idx = t; idx < FF * DD; idx += 128)
        sW[idx] = W1[idx];
    __syncthreads();

    const int wave  = t >> 5;
    const int lane  = t & 31;
    const int half  = lane >> 4;   // 0: lanes 0-15 (K even pair), 1: lanes 16-31
    const int l     = lane & 15;
    const int ncol0 = wave * 16;

    v8f_t c = {0.f, 0.f, 0.f, 0.f, 0.f, 0.f, 0.f, 0.f};
    #pragma unroll
    for (int k0 = 0; k0 < FF; k0 += 4) {
        const int ka = k0 + 2 * half;           // ISA A/B layout: lanes 0-15 hold
        v2f_t a, b;                              // K=k0,k0+1; lanes 16-31 K=k0+2,k0+3
        a.x = sA[l * FF + ka];
        a.y = sA[l * FF + ka + 1];
        b.x = sW[ka * DD + ncol0 + l];
        b.y = sW[(ka + 1) * DD + ncol0 + l];
        c = wmma_f32_4(a, b, c);
    }

    // C/D layout: VGPR r holds row M=r (lanes 0-15) / M=r+8 (lanes 16-31), N=l
    const int   col  = ncol0 + l;
    const float bias = b1[col];
    #pragma unroll
    for (int r = 0; r < 8; ++r) {
        const int row = row0 + half * 8 + r;
        H[(size_t)row * DD + col] = fmaxf(c[r] + bias, 0.f);
    }
}

// ---------------------------------------------------------------------------
// Kernel 2: gather-mean over H rows (all gathers hit L2: H is 25.6 MB).
//   rows [0, B*S):   agg1[b,i] = (sum_j H[s2[b,j,i]] + H[s1[b,i]]) / 33
//   rows [B*S, +B):  aggn[b]   = (sum_j H[s1[b,j]]  + H[batch[b]]) / 33
// One wave per output row; each lane carries a float2 (2 of 64 cols), so each
// gather is one coalesced 256-byte row read.
// ---------------------------------------------------------------------------
__global__ void k2_gather_mean(const float* __restrict__ H,
                               const int* __restrict__ s1,
                               const int* __restrict__ s2,
                               const int* __restrict__ bn,
                               float* __restrict__ agg1,
                               float* __restrict__ aggn) {
    const int wave = threadIdx.x >> 5;
    const int lane = threadIdx.x & 31;
    const int g    = blockIdx.x * 8 + wave;

    float accx = 0.f, accy = 0.f;
    float* dst;

    if (g < BB * SS) {
        const int b = g >> 5;
        const int i = g & 31;
        const int* base = s2 + (size_t)b * SS * SS + i;   // [b, j, i], stride SS
        for (int j = 0; j < SS; ++j) {
            const int    idx = base[j * SS];
            const float2 hv  = *(const float2*)(H + (size_t)idx * DD + 2 * lane);
            accx += hv.x; accy += hv.y;
        }
        {   // self term: h(s1[b,i])
            const int    idx = s1[(size_t)b * SS + i];
            const float2 hv  = *(const float2*)(H + (size_t)idx * DD + 2 * lane);
            accx += hv.x; accy += hv.y;
        }
        dst = agg1 + (size_t)g * DD;
    } else {
        const int b = g - BB * SS;
        for (int j = 0; j < SS; ++j) {
            const int    idx = s1[(size_t)b * SS + j];
            const float2 hv  = *(const float2*)(H + (size_t)idx * DD + 2 * lane);
            accx += hv.x; accy += hv.y;
        }
        {   // self term: h(batch_nodes[b])
            const int    idx = bn[b];
            const float2 hv  = *(const float2*)(H + (size_t)idx * DD + 2 * lane);
            accx += hv.x; accy += hv.y;
        }
        dst = aggn + (size_t)b * DD;
    }

    const float sc = 1.f / 33.f;
    ((float2*)dst)[lane] = make_float2(accx * sc, accy * sc);
}

// ---------------------------------------------------------------------------
// Kernel 3: out[b] = mean over 33 rows of relu(X @ W2 + b2), where
// X = [agg1[b,0..31]; aggn[b]] (33 x 64, zero-padded to 48 rows = 3 M-tiles).
// Block = 128 threads (4 waves), one batch element per block; wave w covers
// output columns w*16..w*16+15. Padded rows are masked out of the column sum.
// ---------------------------------------------------------------------------
__global__ void k3_final(const float* __restrict__ agg1,
                         const float* __restrict__ aggn,
                         const float* __restrict__ W2,
                         const float* __restrict__ b2,
                         float* __restrict__ out) {
    __shared__ float sX[48 * DD];   // 12 KB
    __shared__ float sW[DD * DD];   // 16 KB
    const int t = threadIdx.x;
    const int b = blockIdx.x;

    for (int idx = t; idx < 48 * DD; idx += 128) {
        const int row = idx >> 6, col = idx & 63;
        float v = 0.f;
        if (row < SS)       v = agg1[((size_t)b * SS + row) * DD + col];
        else if (row == SS) v = aggn[(size_t)b * DD + col];
        sX[idx] = v;
    }
    for (int idx = t; idx < DD * DD; idx += 128)
        sW[idx] = W2[idx];
    __syncthreads();

    const int wave  = t >> 5;
    const int lane  = t & 31;
    const int half  = lane >> 4;
    const int l     = lane & 15;
    const int ncol0 = wave * 16;
    const float bias = b2[ncol0 + l];

    float colsum = 0.f;
    #pragma unroll
    for (int mt = 0; mt < 3; ++mt) {
        v8f_t c = {0.f, 0.f, 0.f, 0.f, 0.f, 0.f, 0.f, 0.f};
        #pragma unroll
        for (int k0 = 0; k0 < DD; k0 += 4) {
            const int ka = k0 + 2 * half;
            v2f_t a, bb;
            a.x  = sX[(mt * 16 + l) * DD + ka];
            a.y  = sX[(mt * 16 + l) * DD + ka + 1];
            bb.x = sW[ka * DD + ncol0 + l];
            bb.y = sW[(ka + 1) * DD + ncol0 + l];
            c = wmma_f32_4(a, bb, c);
        }
        #pragma unroll
        for (int r = 0; r < 8; ++r) {
            const int row = mt * 16 + half * 8 + r;
            if (row < SS + 1)                       // mask zero-padded rows
                colsum += fmaxf(c[r] + bias, 0.f);
        }
    }

    // Lanes l and l+16 hold the same column (different row halves): combine.
    colsum += __shfl_xor(colsum, 16, 32);
    if (half == 0)
        out[(size_t)b * DD + ncol0 + l] = colsum * (1.f / 33.f);
}

// ---------------------------------------------------------------------------
extern "C" void kernel_launch(void* const* d_in, const int* in_sizes, int n_in,
                              void* d_out, int out_size, void* d_ws, size_t ws_size,
                              hipStream_t stream) {
    (void)in_sizes; (void)n_in; (void)out_size; (void)ws_size;

    const float* features    = (const float*)d_in[0];  // [N, F]
    const int*   batch_nodes = (const int*)  d_in[1];  // [B]
    const int*   s1_neighs   = (const int*)  d_in[2];  // [B, S]
    const int*   s2_neighs   = (const int*)  d_in[3];  // [B, S, S]
    const float* W1          = (const float*)d_in[4];  // [F, D]
    const float* b1          = (const float*)d_in[5];  // [D]
    const float* W2          = (const float*)d_in[6];  // [D, D]
    const float* b2          = (const float*)d_in[7];  // [D]
    float*       out         = (float*)d_out;          // [B, D]

    // Workspace layout (~34.3 MB total)
    float* H    = (float*)d_ws;                 // [N, D]    25.6 MB
    float* agg1 = H    + (size_t)NN * DD;       // [B*S, D]   8.0 MB
    float* aggn = agg1 + (size_t)BB * SS * DD;  // [B, D]     0.25 MB

    k1_compute_h <<<NN / 16,           128, 0, stream>>>(features, W1, b1, H);
    k2_gather_mean<<<(BB * SS + BB) / 8, 256, 0, stream>>>(H, s1_neighs, s2_neighs,
                                                           batch_nodes, agg1, aggn);
    k3_final     <<<BB,                128, 0, stream>>>(agg1, aggn, W2, b2, out);
}